// SoftChamferLoss_41042707481151
// MI455X (gfx1250) — compile-verified
//
#include <hip/hip_runtime.h>
#include <hip/hip_bf16.h>
#include <math.h>

typedef __attribute__((ext_vector_type(2))) float v2f;
typedef __attribute__((ext_vector_type(8))) float v8f;

#define ZETA     0.005f
#define EPS_DIST 0.01f
#define FLT_BIG  3.402823466e38f
#define LOG2E    1.44269504088896340736f
#define NEG_HALF_LOG2E (-0.72134752044448170368f)   // -0.5 * log2(e)

// XOR-swizzle across all 32 lanes with an immediate pattern:
// offset[15]=0 (group-of-32), xor_mask=MASK, or_mask=0, and_mask=0x1F
template <int MASK>
__device__ __forceinline__ float swz_xor(float v) {
    return __int_as_float(
        __builtin_amdgcn_ds_swizzle(__float_as_int(v), (MASK << 10) | 0x1F));
}

// ---------------------------------------------------------------------------
// Init: zero density accumulators, +inf min accumulators, precompute |pc2|^2
// and the pre-scaled s*|pc2|^2 used by the density pass's WMMA B operand.
// ---------------------------------------------------------------------------
__global__ void scl_init_kernel(const float* __restrict__ pc2,
                                float* __restrict__ dens12, float* __restrict__ dens21,
                                unsigned* __restrict__ min1, unsigned* __restrict__ min2,
                                float* __restrict__ n2, float* __restrict__ n2s,
                                int N, int BN) {
    int i = blockIdx.x * blockDim.x + threadIdx.x;
    if (i >= BN) return;
    dens12[i] = 0.f;
    dens21[i] = 0.f;
    min1[i] = 0x7F800000u;  // +inf
    min2[i] = 0x7F800000u;
    int b = i / N, n = i - b * N;
    const float* base = pc2 + (size_t)b * 3 * N;
    float x = base[n], y = base[N + n], z = base[2 * N + n];
    float nn = x * x + y * y + z * z;
    n2[i]  = nn;
    n2s[i] = NEG_HALF_LOG2E * nn;
}

// ---------------------------------------------------------------------------
// Tile pass: one wave computes 16x16 squared-distance tiles with a single
// V_WMMA_F32_16X16X4_F32 per tile:
//   A = (k*x, k*y, k*z, 1), B = (x, y, z, s*|b|^2), C = s*|a|^2, k = -2s
// DENSITY=1: s = -0.5*log2(e) -> D is the exp2 argument; row/col sums.
// DENSITY=0: s = 1            -> D is the squared distance; row/col mins.
// B operand is software-pipelined; the final tile is peeled so the in-loop
// prefetch is a pure constant-stride pointer bump (no clamp/cndmask).
// ---------------------------------------------------------------------------
template <int DENSITY>
__global__ __launch_bounds__(256)
void scl_tile_kernel(const float* __restrict__ A_cloud,  // p1 or p1_warp, [B,3,N]
                     const float* __restrict__ p2,       // [B,3,N]
                     const float* __restrict__ n2any,    // [B,N] |b|^2 (pre-scaled if DENSITY)
                     float* __restrict__ rowOut,         // dens12 / min1(bits)
                     float* __restrict__ colOut,         // dens21 / min2(bits)
                     int N) {
    const int b    = blockIdx.y;
    const int trb  = blockIdx.x * 16;          // row-tile base
    const int lane = threadIdx.x & 31;
    const int wave = threadIdx.x >> 5;
    const int lm   = lane & 15;
    const int hi   = lane >> 4;                // 0: lanes 0-15, 1: lanes 16-31

    const float* ax = A_cloud + (size_t)b * 3 * N;
    const float* ay = ax + N;
    const float* az = ay + N;
    const float* bx = p2 + (size_t)b * 3 * N;
    const float* by = bx + N;
    const float* bz = by + N;
    const float* bn = n2any + (size_t)b * N;   // already s-scaled for DENSITY

    const float ka = DENSITY ? LOG2E : -2.0f;           // A coord scale (-2s)
    const float ks = DENSITY ? NEG_HALF_LOG2E : 1.0f;   // s (C scale)

    // A operand (16x4 f32): lanes<16 carry K={0,1}, lanes>=16 carry K={2,3}
    const int arow = trb + lm;
    v2f a;
    if (!hi) { a.x = ka * ax[arow]; a.y = ka * ay[arow]; }
    else     { a.x = ka * az[arow]; a.y = 1.0f; }

    // C operand: s * |a_row|^2 for rows trb + v + 8*hi (matches D layout)
    v8f c;
#pragma unroll
    for (int v = 0; v < 8; ++v) {
        int r = trb + v + 8 * hi;
        float x = ax[r], y = ay[r], z = az[r];
        c[v] = ks * (x * x + y * y + z * z);
    }

    float rowAcc[8];
#pragma unroll
    for (int v = 0; v < 8; ++v) rowAcc[v] = DENSITY ? 0.f : FLT_BIG;

    float* const colBase = colOut + (size_t)b * N;

    // per-tile processing: WMMA + row/col accumulation for column index `col`
    auto tile_process = [&](v2f bv, int col) {
        v8f d = __builtin_amdgcn_wmma_f32_16x16x4_f32(
            false, a, false, bv, (short)0, c, false, false);
        if (DENSITY) {
            float g[8];
#pragma unroll
            for (int v = 0; v < 8; ++v) {
                g[v] = __builtin_amdgcn_exp2f(d[v]);   // D already scaled
                rowAcc[v] += g[v];
            }
            float cs01 = g[0] + g[1], cs23 = g[2] + g[3];
            float cs45 = g[4] + g[5], cs67 = g[6] + g[7];
            float cs = (cs01 + cs23) + (cs45 + cs67);
            cs += swz_xor<16>(cs);  // combine both row-halves of the column
            if (!hi)
                atomicAdd(colBase + col, cs);
        } else {
            float m01 = fminf(d[0], d[1]), m23 = fminf(d[2], d[3]);
            float m45 = fminf(d[4], d[5]), m67 = fminf(d[6], d[7]);
#pragma unroll
            for (int v = 0; v < 8; ++v)
                rowAcc[v] = fminf(rowAcc[v], d[v]);
            float cm = fminf(fminf(m01, m23), fminf(m45, m67));
            cm = fminf(cm, swz_xor<16>(cm));
            if (!hi)  // clamp >=0 so uint bit-order == float order
                atomicMin((unsigned*)colBase + col,
                          __float_as_uint(fmaxf(cm, 0.f)));
        }
    };

    // Branch-free per-half B sources: lanes<16 read (x,y), lanes>=16 read (z, s|b|^2)
    const float* pXa = (hi ? bz : bx) + wave * 16 + lm;
    const float* pYa = (hi ? bn : by) + wave * 16 + lm;

    const int numCT = N >> 4;
    if (wave < numCT) {
        int col = wave * 16 + lm;
        v2f bv;
        bv.x = *pXa;
        bv.y = *pYa;

        // main loop: every iteration has a successor -> prefetch is a pure
        // +128-float pointer bump, fully strength-reduced
        int ct = wave;
        for (; ct + 8 < numCT; ct += 8) {
            pXa += 128;
            pYa += 128;
            v2f bnext;
            bnext.x = *pXa;
            bnext.y = *pYa;

            tile_process(bv, col);

            bv = bnext;
            col += 128;
        }
        // peeled final tile
        tile_process(bv, col);
    }

    // Row reduction across the 16 lanes of each half-wave (immediate swizzles)
#pragma unroll
    for (int v = 0; v < 8; ++v) {
        float r = rowAcc[v];
        if (DENSITY) {
            r += swz_xor<1>(r);
            r += swz_xor<2>(r);
            r += swz_xor<4>(r);
            r += swz_xor<8>(r);
        } else {
            r = fminf(r, swz_xor<1>(r));
            r = fminf(r, swz_xor<2>(r));
            r = fminf(r, swz_xor<4>(r));
            r = fminf(r, swz_xor<8>(r));
        }
        if (lm == 0) {
            size_t idx = (size_t)b * N + trb + v + 8 * hi;
            if (DENSITY) atomicAdd(rowOut + idx, r);
            else         atomicMin((unsigned*)rowOut + idx,
                                   __float_as_uint(fmaxf(r, 0.f)));
        }
    }
}

// ---------------------------------------------------------------------------
// Finalize: masks from densities, relu(min - eps), means, sum -> scalar
// ---------------------------------------------------------------------------
__global__ __launch_bounds__(256)
void scl_finalize_kernel(const float* __restrict__ dens12, const float* __restrict__ dens21,
                         const unsigned* __restrict__ min1, const unsigned* __restrict__ min2,
                         float* __restrict__ out, int N, int BN) {
    __shared__ float s1[256], s2[256];
    const float dscale = 1.0f / (2.5f * (float)N);  // mean of exp / (2.5*bw), bw=1
    float a1 = 0.f, a2 = 0.f;
    for (int i = threadIdx.x; i < BN; i += 256) {
        float m1 = (dens12[i] * dscale > ZETA) ? 1.f : 0.f;
        float m2 = (dens21[i] * dscale > ZETA) ? 1.f : 0.f;
        a1 += fmaxf(__uint_as_float(min1[i]) - EPS_DIST, 0.f) * m1;
        a2 += fmaxf(__uint_as_float(min2[i]) - EPS_DIST, 0.f) * m2;
    }
    s1[threadIdx.x] = a1;
    s2[threadIdx.x] = a2;
    __syncthreads();
    for (int s = 128; s > 0; s >>= 1) {
        if (threadIdx.x < s) {
            s1[threadIdx.x] += s1[threadIdx.x + s];
            s2[threadIdx.x] += s2[threadIdx.x + s];
        }
        __syncthreads();
    }
    if (threadIdx.x == 0) out[0] = (s1[0] + s2[0]) / (float)BN;
}

// ---------------------------------------------------------------------------
extern "C" void kernel_launch(void* const* d_in, const int* in_sizes, int n_in,
                              void* d_out, int out_size, void* d_ws, size_t ws_size,
                              hipStream_t stream) {
    const float* pc1  = (const float*)d_in[0];  // [B,3,N]
    const float* pc2  = (const float*)d_in[1];  // [B,3,N]
    const float* pc1w = (const float*)d_in[2];  // [B,3,N]

    const int B  = 4;
    const int N  = in_sizes[0] / (3 * B);       // 4096
    const int BN = B * N;

    float*    ws     = (float*)d_ws;
    float*    dens12 = ws;                       // [B,N]
    float*    dens21 = ws + BN;                  // [B,N]
    unsigned* min1   = (unsigned*)(ws + 2 * BN); // [B,N] float bits
    unsigned* min2   = (unsigned*)(ws + 3 * BN); // [B,N] float bits
    float*    n2     = ws + 4 * BN;              // [B,N] |pc2|^2
    float*    n2s    = ws + 5 * BN;              // [B,N] s*|pc2|^2 (density pass)

    scl_init_kernel<<<(BN + 255) / 256, 256, 0, stream>>>(
        pc2, dens12, dens21, min1, min2, n2, n2s, N, BN);

    dim3 grid(N / 16, B);
    // Density pass: p1 vs p2 (row sums -> dens12, col sums -> dens21)
    scl_tile_kernel<1><<<grid, 256, 0, stream>>>(pc1, pc2, n2s, dens12, dens21, N);
    // NN pass: p1_warp vs p2 (row mins -> dist1, col mins -> dist2)
    scl_tile_kernel<0><<<grid, 256, 0, stream>>>(pc1w, pc2, n2,
                                                 (float*)min1, (float*)min2, N);

    scl_finalize_kernel<<<1, 256, 0, stream>>>(dens12, dens21, min1, min2,
                                               (float*)d_out, N, BN);
}